// MasifLigandNet_71305047048705
// MI455X (gfx1250) — compile-verified
//
#include <hip/hip_runtime.h>

#define B_DIM   64
#define N_DIM   10000
#define L_DIM   50
#define D_DIM   512
#define O_DIM   7
#define K_NEIGH 10
#define LN_EPS  1e-5f

typedef __attribute__((ext_vector_type(2))) float v2f;
typedef __attribute__((ext_vector_type(8))) float v8f;

// ---------------------------------------------------------------- zero mask
__global__ void zero_mask_kernel(unsigned* __restrict__ mask, int n) {
    int i = blockIdx.x * blockDim.x + threadIdx.x;
    if (i < n) mask[i] = 0u;
}

// ---------------------------------------------------------------- top-k mask
// One wave32 per (batch, ligand atom). Each lane keeps a register-resident
// sorted top-10 over its strided slice of N, then 10 rounds of wave-wide
// 64-bit-key min-reduction extract the global 10 nearest; lane 0 sets mask.
__global__ __launch_bounds__(256) void topk_mask_kernel(
        const float* __restrict__ pos,
        const float* __restrict__ lig,
        unsigned* __restrict__ mask) {
    const int wave = blockIdx.x * 8 + (threadIdx.x >> 5);
    const int lane = threadIdx.x & 31;
    const int b = wave / L_DIM;
    const int l = wave % L_DIM;

    const float* lg = lig + ((size_t)b * L_DIM + l) * 3;
    const float lx = lg[0], ly = lg[1], lz = lg[2];
    const float* pb = pos + (size_t)b * N_DIM * 3;

    float d[K_NEIGH];
    int   id[K_NEIGH];
#pragma unroll
    for (int t = 0; t < K_NEIGH; ++t) { d[t] = 3.4e38f; id[t] = 0; }

    for (int i = lane; i < N_DIM; i += 32) {
        const float dx = pb[i * 3 + 0] - lx;
        const float dy = pb[i * 3 + 1] - ly;
        const float dz = pb[i * 3 + 2] - lz;
        const float dd = dx * dx + dy * dy + dz * dz;
        if (dd < d[K_NEIGH - 1]) {
            // branch-free sorted insert (constant indices after unroll ->
            // arrays stay in VGPRs, no scratch)
#pragma unroll
            for (int t = K_NEIGH - 1; t >= 0; --t) {
                const float prevd  = (t > 0) ? d[t - 1]  : -1.0f;
                const int   previd = (t > 0) ? id[t - 1] : 0;
                if (dd < d[t]) {
                    if (dd < prevd) { d[t] = prevd; id[t] = previd; }
                    else            { d[t] = dd;    id[t] = i;      }
                }
            }
        }
    }

    // Wave-wide merge: extract 10 global minima. d2 >= 0 so float bit
    // pattern is order-monotonic; key = (dist_bits << 32) | vertex_idx.
    for (int r = 0; r < K_NEIGH; ++r) {
        const unsigned long long mykey =
            (((unsigned long long)__float_as_uint(d[0])) << 32) | (unsigned)id[0];
        unsigned long long key = mykey;
#pragma unroll
        for (int off = 16; off > 0; off >>= 1) {
            unsigned long long o = __shfl_xor(key, off, 32);
            key = (o < key) ? o : key;
        }
        if (mykey == key) {  // winning lane pops its head (constant-index shift)
#pragma unroll
            for (int t = 0; t < K_NEIGH - 1; ++t) { d[t] = d[t + 1]; id[t] = id[t + 1]; }
            d[K_NEIGH - 1] = 3.4e38f; id[K_NEIGH - 1] = 0;
        }
        if (lane == 0) {
            const int widx = (int)(unsigned)(key & 0xffffffffu);
            mask[(size_t)b * N_DIM + widx] = 1u;  // benign racing writes of 1
        }
    }
}

// ---------------------------------------------------------------- gather pool
// One block per batch: compact mask -> index list in LDS, then gather only
// the <=500 selected rows of x (64 MB total vs 1.31 GB for the dense mean).
__global__ __launch_bounds__(256) void pool_kernel(
        const float* __restrict__ x,
        const unsigned* __restrict__ mask,
        float* __restrict__ emb) {
    __shared__ int idxs[512];
    __shared__ int cnt;
    const int b = blockIdx.x;
    const int tid = threadIdx.x;
    if (tid == 0) cnt = 0;
    __syncthreads();

    const unsigned* mb = mask + (size_t)b * N_DIM;
    for (int i = tid; i < N_DIM; i += 256) {
        if (mb[i]) {
            int p = atomicAdd(&cnt, 1);
            if (p < 512) idxs[p] = i;      // unique count <= L*K = 500
        }
    }
    __syncthreads();

    const int n = cnt < 512 ? cnt : 512;
    const float* xb = x + (size_t)b * N_DIM * D_DIM;
    float a0 = 0.f, a1 = 0.f;
    for (int r = 0; r < n; ++r) {
        const float* row = xb + (size_t)idxs[r] * D_DIM;
        if (r + 1 < n)  // emits global_prefetch_b8 on gfx1250
            __builtin_prefetch(xb + (size_t)idxs[r + 1] * D_DIM + tid, 0, 1);
        a0 += row[tid];
        a1 += row[tid + 256];
    }
    const float inv = 1.0f / (float)n;
    emb[(size_t)b * D_DIM + tid]       = a0 * inv;
    emb[(size_t)b * D_DIM + tid + 256] = a1 * inv;
}

// ---------------------------------------------------------------- WMMA GEMM
// h = emb[64,512] @ W1[512,512] + b1, fp32 via V_WMMA_F32_16X16X4_F32.
// One wave per 16x16 output tile: 4 M-tiles x 32 N-tiles = 128 waves,
// K-loop of 128 steps of 4.
__global__ __launch_bounds__(256) void gemm1_wmma_kernel(
        const float* __restrict__ emb,
        const float* __restrict__ W1,
        const float* __restrict__ b1,
        float* __restrict__ h) {
    const int tile = blockIdx.x * 8 + (threadIdx.x >> 5);  // 0..127
    const int lane = threadIdx.x & 31;
    const int mt = tile >> 5;     // 0..3
    const int nt = tile & 31;     // 0..31
    const int half = lane >> 4;   // lanes 16-31 hold K+2/K+3 (A) / rows M+8 (C/D)
    const int l15  = lane & 15;
    const int mbase = mt * 16;
    const int nbase = nt * 16;
    const int col = nbase + l15;

    v8f c = {};
    const float* arow = emb + (size_t)(mbase + l15) * D_DIM;
    for (int k0 = 0; k0 < D_DIM; k0 += 4) {
        const int kk = k0 + half * 2;
        v2f a;
        a.x = arow[kk];
        a.y = arow[kk + 1];
        v2f bf;
        bf.x = W1[(size_t)kk * D_DIM + col];
        bf.y = W1[(size_t)(kk + 1) * D_DIM + col];
        c = __builtin_amdgcn_wmma_f32_16x16x4_f32(
                /*neg_a=*/false, a, /*neg_b=*/false, bf,
                /*c_mod=*/(short)0, c, /*reuse_a=*/false, /*reuse_b=*/false);
    }

    const float bias = b1[col];
    const int rowbase = mbase + half * 8;   // C/D layout: VGPR r -> M=r (+8 hi half)
#pragma unroll
    for (int r = 0; r < 8; ++r)
        h[(size_t)(rowbase + r) * D_DIM + col] = c[r] + bias;
}

// ---------------------------------------------------------------- LN+SiLU+out
__global__ __launch_bounds__(256) void head_kernel(
        const float* __restrict__ h,
        const float* __restrict__ gamma,
        const float* __restrict__ beta,
        const float* __restrict__ W2,
        const float* __restrict__ b2,
        float* __restrict__ out) {
    __shared__ float red[256];
    __shared__ float s[D_DIM];
    const int b = blockIdx.x;
    const int tid = threadIdx.x;

    const float v0 = h[(size_t)b * D_DIM + tid];
    const float v1 = h[(size_t)b * D_DIM + tid + 256];

    red[tid] = v0 + v1;
    __syncthreads();
    for (int off = 128; off > 0; off >>= 1) {
        if (tid < off) red[tid] += red[tid + off];
        __syncthreads();
    }
    const float mu = red[0] * (1.0f / D_DIM);
    __syncthreads();

    red[tid] = (v0 - mu) * (v0 - mu) + (v1 - mu) * (v1 - mu);
    __syncthreads();
    for (int off = 128; off > 0; off >>= 1) {
        if (tid < off) red[tid] += red[tid + off];
        __syncthreads();
    }
    const float rstd = rsqrtf(red[0] * (1.0f / D_DIM) + LN_EPS);

    float a0 = (v0 - mu) * rstd * gamma[tid] + beta[tid];
    float a1 = (v1 - mu) * rstd * gamma[tid + 256] + beta[tid + 256];
    a0 = a0 / (1.0f + __expf(-a0));   // SiLU
    a1 = a1 / (1.0f + __expf(-a1));
    s[tid] = a0;
    s[tid + 256] = a1;
    __syncthreads();

    if (tid < O_DIM) {
        float acc = b2[tid];
        for (int dd = 0; dd < D_DIM; ++dd)
            acc += s[dd] * W2[(size_t)dd * O_DIM + tid];
        out[(size_t)b * O_DIM + tid] = acc;
    }
}

// ---------------------------------------------------------------- launch
extern "C" void kernel_launch(void* const* d_in, const int* in_sizes, int n_in,
                              void* d_out, int out_size, void* d_ws, size_t ws_size,
                              hipStream_t stream) {
    const float* pos   = (const float*)d_in[0];
    const float* x     = (const float*)d_in[1];
    const float* lig   = (const float*)d_in[2];
    const float* W1    = (const float*)d_in[3];
    const float* b1    = (const float*)d_in[4];
    const float* gamma = (const float*)d_in[5];
    const float* beta  = (const float*)d_in[6];
    const float* W2    = (const float*)d_in[7];
    const float* b2    = (const float*)d_in[8];
    float* out = (float*)d_out;

    char* ws = (char*)d_ws;
    unsigned* mask = (unsigned*)ws;                                    // B*N u32
    float* emb = (float*)(ws + (size_t)B_DIM * N_DIM * sizeof(unsigned)); // B*D f32
    float* h   = emb + (size_t)B_DIM * D_DIM;                          // B*D f32

    const int maskN = B_DIM * N_DIM;
    zero_mask_kernel<<<(maskN + 255) / 256, 256, 0, stream>>>(mask, maskN);
    topk_mask_kernel<<<(B_DIM * L_DIM) / 8, 256, 0, stream>>>(pos, lig, mask);
    pool_kernel<<<B_DIM, 256, 0, stream>>>(x, mask, emb);
    gemm1_wmma_kernel<<<(4 * 32) / 8, 256, 0, stream>>>(emb, W1, b1, h);
    head_kernel<<<B_DIM, 256, 0, stream>>>(h, gamma, beta, W2, b2, out);
}